// TransLayer_32212254720238
// MI455X (gfx1250) — compile-verified
//
#include <hip/hip_runtime.h>
#include <math.h>

#define HEADS   8
#define DH      64
#define NSEQ    4096
#define BATCH   4
#define BH      (BATCH*HEADS)        // 32
#define LM      256
#define LSZ     16                   // NSEQ / LM
#define DIM     512
#define INNER   (HEADS*DH)           // 512
#define ROWS    (BATCH*NSEQ)         // 16384
#define KW      33
#define QSCALE  0.125f               // 64^-0.5

typedef __attribute__((ext_vector_type(2))) float v2f;
typedef __attribute__((ext_vector_type(8))) float v8f;

// ---------------------------------------------------------------------------
// Wave-level 64x64 fp32 tile GEMM using V_WMMA_F32_16X16X4_F32 (4x4 register
// blocking: 16 independent accumulators, 4 A-frags + 4 B-frags per K-step ->
// 16 WMMAs per 8 b64 loads = 16 FLOP/byte out of L1/L2).
// A: MxK row-major (lda); B: KxN row-major (ldb) if BT==false,
//                         B: NxK row-major (ldb, i.e. A@B^T) if BT==true.
// Frag layouts per CDNA5 ISA 7.12.2:
//   A 16x4 : lanes 0-15 -> M=lane, VGPR0/1 = K={k,k+1}; lanes 16-31 -> K={k+2,k+3}
//   B 4x16 : VGPR0 lanes 0-15 -> K=k,   N=lane; lanes 16-31 -> K=k+2
//            VGPR1 lanes 0-15 -> K=k+1;          lanes 16-31 -> K=k+3
//   D 16x16: VGPR r, lanes 0-15 -> M=r, N=lane; lanes 16-31 -> M=r+8
// ---------------------------------------------------------------------------
#define WMMA_F32(a, b, c) \
  __builtin_amdgcn_wmma_f32_16x16x4_f32(false, (a), false, (b), (short)0, (c), false, false)

template<bool BT>
__device__ __forceinline__ void wave_tile_gemm64(const float* __restrict__ A, int lda,
                                                 const float* __restrict__ B, int ldb,
                                                 int m0, int n0, int K, v8f c[4][4]) {
  const int lane = threadIdx.x & 31;
  const int l15  = lane & 15;
  const int hi   = lane >> 4;
  const float* __restrict__ ar[4];
  const float* __restrict__ br[4];
#pragma unroll
  for (int i = 0; i < 4; ++i) ar[i] = A + (size_t)(m0 + 16 * i + l15) * lda;
  if (BT) {
#pragma unroll
    for (int j = 0; j < 4; ++j) br[j] = B + (size_t)(n0 + 16 * j + l15) * ldb;
  }
#pragma unroll 2
  for (int k0 = 0; k0 < K; k0 += 4) {
    const int kk = k0 + 2 * hi;
    v2f a[4], b[4];
#pragma unroll
    for (int i = 0; i < 4; ++i) { a[i].x = ar[i][kk]; a[i].y = ar[i][kk + 1]; }
    if (BT) {                       // B^T[k][n] = B[n][k]
#pragma unroll
      for (int j = 0; j < 4; ++j) { b[j].x = br[j][kk]; b[j].y = br[j][kk + 1]; }
    } else {                        // B[k][n]
      const float* bp = B + (size_t)kk * ldb + (n0 + l15);
#pragma unroll
      for (int j = 0; j < 4; ++j) { b[j].x = bp[16 * j]; b[j].y = bp[ldb + 16 * j]; }
    }
#pragma unroll
    for (int i = 0; i < 4; ++i)
#pragma unroll
      for (int j = 0; j < 4; ++j)
        c[i][j] = WMMA_F32(a[i], b[j], c[i][j]);
  }
}

#define ACC_INIT64                                     \
  v8f c[4][4];                                         \
  _Pragma("unroll")                                    \
  for (int i_ = 0; i_ < 4; ++i_)                       \
    _Pragma("unroll")                                  \
    for (int j_ = 0; j_ < 4; ++j_)                     \
      c[i_][j_] = (v8f){0.f,0.f,0.f,0.f,0.f,0.f,0.f,0.f}

// ---------------------------------------------------------------------------
// LayerNorm: one block per row of 512
// ---------------------------------------------------------------------------
__global__ void ln_kernel(const float* __restrict__ x, const float* __restrict__ g,
                          const float* __restrict__ be, float* __restrict__ xn) {
  __shared__ float red[256];
  const size_t row = blockIdx.x;
  const float* xr = x + row * DIM;
  const int t = threadIdx.x;
  float a0 = xr[t], a1 = xr[t + 256];
  red[t] = a0 + a1;
  __syncthreads();
  for (int s = 128; s > 0; s >>= 1) { if (t < s) red[t] += red[t + s]; __syncthreads(); }
  const float mu = red[0] * (1.f / DIM);
  __syncthreads();
  const float d0 = a0 - mu, d1 = a1 - mu;
  red[t] = d0 * d0 + d1 * d1;
  __syncthreads();
  for (int s = 128; s > 0; s >>= 1) { if (t < s) red[t] += red[t + s]; __syncthreads(); }
  const float rs = rsqrtf(red[0] * (1.f / DIM) + 1e-5f);
  xn[row * DIM + t]       = d0 * rs * g[t]       + be[t];
  xn[row * DIM + t + 256] = d1 * rs * g[t + 256] + be[t + 256];
}

// ---------------------------------------------------------------------------
// QKV GEMM: [16384,512] @ w_qkv[1536,512]^T, epilogue scatters to
// q/k/v in [b,h,n,d] layout, scaling q by dh^-0.5.
// 64-wide tile aligned to 64 => sel and head are constant per tile.
// ---------------------------------------------------------------------------
__global__ void __launch_bounds__(256, 1)
qkv_gemm_kernel(const float* __restrict__ xn, const float* __restrict__ wqkv,
                float* __restrict__ q, float* __restrict__ k,
                float* __restrict__ v) {
  const int wave = threadIdx.x >> 5;
  const long long tile = (long long)blockIdx.x * 8 + wave;
  const int tilesN = (3 * INNER) / 64;           // 24
  if (tile >= (long long)(ROWS / 64) * tilesN) return;
  const int ty = (int)(tile / tilesN);
  const int tx = (int)(tile % tilesN);
  ACC_INIT64;
  wave_tile_gemm64<true>(xn, DIM, wqkv, DIM, ty * 64, tx * 64, DIM, c);
  const int lane = threadIdx.x & 31, l15 = lane & 15, hi = lane >> 4;
  const int colBase = tx * 64;
  const int sel = colBase >> 9;                  // 0:q 1:k 2:v
  const int h   = (colBase & 511) >> 6;          // head (64-aligned tile)
  float* dst = (sel == 0) ? q : ((sel == 1) ? k : v);
  const float sc = (sel == 0) ? QSCALE : 1.0f;
#pragma unroll
  for (int i = 0; i < 4; ++i) {
#pragma unroll
    for (int rr = 0; rr < 8; ++rr) {
      const int idx = ty * 64 + 16 * i + rr + 8 * hi;   // idx = b*NSEQ + n
      const int bb = idx >> 12, nn = idx & (NSEQ - 1);
      const size_t base = ((size_t)(bb * HEADS + h) * NSEQ + nn) * DH + l15;
#pragma unroll
      for (int j = 0; j < 4; ++j)
        dst[base + 16 * j] = c[i][j][rr] * sc;
    }
  }
}

// ---------------------------------------------------------------------------
// Landmark mean pooling: ql/kl[bh,m,d] = mean_l q/k[bh, m*16+l, d]
// ---------------------------------------------------------------------------
__global__ void landmark_kernel(const float* __restrict__ q, const float* __restrict__ k,
                                float* __restrict__ ql, float* __restrict__ kl) {
  const int d = threadIdx.x;                     // 64 threads
  const long long idx = blockIdx.x;              // bh*LM + m
  const long long bh = idx >> 8;
  const int m = (int)(idx & (LM - 1));
  const float* qb = q + ((bh * NSEQ) + (size_t)m * LSZ) * DH + d;
  const float* kb = k + ((bh * NSEQ) + (size_t)m * LSZ) * DH + d;
  float sq = 0.f, sk = 0.f;
#pragma unroll
  for (int l = 0; l < LSZ; ++l) { sq += qb[l * DH]; sk += kb[l * DH]; }
  ql[(bh * LM + m) * DH + d] = sq * (1.f / LSZ);
  kl[(bh * LM + m) * DH + d] = sk * (1.f / LSZ);
}

// ---------------------------------------------------------------------------
// Batched GEMMs (one 64x64 tile per wave, 8 waves per block, batch = blockIdx.y)
// ---------------------------------------------------------------------------
__global__ void __launch_bounds__(256, 1)
gemm_abt_kernel(const float* __restrict__ A, const float* __restrict__ B,
                float* __restrict__ C, int M, int N, int K,
                long long sA, long long sB, long long sC) {
  const int wave = threadIdx.x >> 5;
  const long long tile = (long long)blockIdx.x * 8 + wave;
  const int tilesN = N >> 6;
  if (tile >= (long long)(M >> 6) * tilesN) return;
  const int ty = (int)(tile / tilesN);
  const int tx = (int)(tile % tilesN);
  const float* Ab = A + (size_t)blockIdx.y * sA;
  const float* Bb = B + (size_t)blockIdx.y * sB;
  float* Cb = C + (size_t)blockIdx.y * sC;
  ACC_INIT64;
  wave_tile_gemm64<true>(Ab, K, Bb, K, ty * 64, tx * 64, K, c);
  const int lane = threadIdx.x & 31, l15 = lane & 15, hi = lane >> 4;
#pragma unroll
  for (int i = 0; i < 4; ++i) {
#pragma unroll
    for (int rr = 0; rr < 8; ++rr) {
      float* cr = Cb + (size_t)(ty * 64 + 16 * i + rr + 8 * hi) * N + tx * 64 + l15;
#pragma unroll
      for (int j = 0; j < 4; ++j) cr[16 * j] = c[i][j][rr];
    }
  }
}

// C = beta*I + alpha*(A @ B), A: MxK, B: KxN, all row-major
__global__ void __launch_bounds__(256, 1)
gemm_nn_kernel(const float* __restrict__ A, const float* __restrict__ B,
               float* __restrict__ C, int M, int N, int K,
               long long sA, long long sB, long long sC,
               float alpha, float beta_diag) {
  const int wave = threadIdx.x >> 5;
  const long long tile = (long long)blockIdx.x * 8 + wave;
  const int tilesN = N >> 6;
  if (tile >= (long long)(M >> 6) * tilesN) return;
  const int ty = (int)(tile / tilesN);
  const int tx = (int)(tile % tilesN);
  const float* Ab = A + (size_t)blockIdx.y * sA;
  const float* Bb = B + (size_t)blockIdx.y * sB;
  float* Cb = C + (size_t)blockIdx.y * sC;
  ACC_INIT64;
  wave_tile_gemm64<false>(Ab, K, Bb, N, ty * 64, tx * 64, K, c);
  const int lane = threadIdx.x & 31, l15 = lane & 15, hi = lane >> 4;
#pragma unroll
  for (int i = 0; i < 4; ++i) {
#pragma unroll
    for (int rr = 0; rr < 8; ++rr) {
      const int row = ty * 64 + 16 * i + rr + 8 * hi;
      float* cr = Cb + (size_t)row * N + tx * 64 + l15;
#pragma unroll
      for (int j = 0; j < 4; ++j) {
        const int col = tx * 64 + l15 + 16 * j;
        float val = alpha * c[i][j][rr];
        if (row == col) val += beta_diag;
        cr[16 * j] = val;
      }
    }
  }
}

// ---------------------------------------------------------------------------
// Row softmax in place; cols in {256, 4096}; one block per row, row cached in LDS
// ---------------------------------------------------------------------------
__global__ void softmax_rows_kernel(float* __restrict__ p, int cols) {
  __shared__ float buf[4096];
  __shared__ float red[256];
  const size_t row = blockIdx.x;
  float* pr = p + row * (size_t)cols;
  const int t = threadIdx.x;
  const int per = cols >> 8;
  float mx = -3.4e38f;
  for (int i = 0; i < per; ++i) {
    const float v = pr[t + i * 256];
    buf[t + i * 256] = v;
    mx = fmaxf(mx, v);
  }
  red[t] = mx; __syncthreads();
  for (int s = 128; s > 0; s >>= 1) { if (t < s) red[t] = fmaxf(red[t], red[t + s]); __syncthreads(); }
  const float m = red[0]; __syncthreads();
  float sum = 0.f;
  for (int i = 0; i < per; ++i) {
    const float e = __expf(buf[t + i * 256] - m);
    buf[t + i * 256] = e;
    sum += e;
  }
  red[t] = sum; __syncthreads();
  for (int s = 128; s > 0; s >>= 1) { if (t < s) red[t] += red[t + s]; __syncthreads(); }
  const float inv = 1.f / red[0];
  for (int i = 0; i < per; ++i) pr[t + i * 256] = buf[t + i * 256] * inv;
}

// ---------------------------------------------------------------------------
// Moore-Penrose pinv scaling: global max row-sum and col-sum of attn2
// ---------------------------------------------------------------------------
__global__ void pinv_scale_partial(const float* __restrict__ a2, float* __restrict__ part) {
  __shared__ float rc[256], rr[256];
  const int bh = blockIdx.x;
  const int j = threadIdx.x;
  const float* a = a2 + (size_t)bh * LM * LM;
  float cs = 0.f, rs = 0.f;
  for (int i = 0; i < LM; ++i) { cs += a[(size_t)i * LM + j]; rs += a[(size_t)j * LM + i]; }
  rc[j] = cs; rr[j] = rs; __syncthreads();
  for (int s = 128; s > 0; s >>= 1) {
    if (j < s) { rc[j] = fmaxf(rc[j], rc[j + s]); rr[j] = fmaxf(rr[j], rr[j + s]); }
    __syncthreads();
  }
  if (j == 0) { part[bh] = rc[0]; part[BH + bh] = rr[0]; }
}

__global__ void pinv_scale_final(const float* __restrict__ part, float* __restrict__ scal) {
  if (threadIdx.x == 0) {
    float mc = 0.f, mr = 0.f;
    for (int i = 0; i < BH; ++i) { mc = fmaxf(mc, part[i]); mr = fmaxf(mr, part[BH + i]); }
    scal[0] = 1.f / (mc * mr);
  }
}

__global__ void pinv_init_kernel(const float* __restrict__ a2, const float* __restrict__ scal,
                                 float* __restrict__ z) {
  const size_t idx = (size_t)blockIdx.x * 256 + threadIdx.x;   // BH*LM*LM total
  const size_t bh = idx >> 16;
  const int ij = (int)(idx & 65535);
  const int i = ij >> 8, j = ij & 255;
  z[idx] = a2[(bh << 16) + ((size_t)j << 8) + i] * scal[0];
}

// out = cdiag*I - in   (elementwise over batched 256x256)
__global__ void axpy_diag_kernel(const float* __restrict__ in, float* __restrict__ out,
                                 float cdiag) {
  const size_t idx = (size_t)blockIdx.x * 256 + threadIdx.x;
  const int ij = (int)(idx & 65535);
  const int i = ij >> 8, j = ij & 255;
  out[idx] = ((i == j) ? cdiag : 0.f) - in[idx];
}

// ---------------------------------------------------------------------------
// Depthwise conv residual (kernel 33 along seq, per head) + transpose heads back:
// outc[b, n, h*64+d] = outh[bh, n, d] + sum_t conv_w[h, t] * v[bh, n+t-16, d]
// ---------------------------------------------------------------------------
__global__ void conv_add_transpose_kernel(const float* __restrict__ outh,
                                          const float* __restrict__ v,
                                          const float* __restrict__ cw,
                                          float* __restrict__ outc) {
  const int sub = threadIdx.x >> 6;              // 4 seq positions per block
  const int d = threadIdx.x & 63;
  const size_t idx = (size_t)blockIdx.x * 4 + sub;   // bh*NSEQ + n
  const size_t bh = idx >> 12;
  const int n = (int)(idx & (NSEQ - 1));
  const int b = (int)(bh >> 3), h = (int)(bh & 7);
  float acc = outh[(bh * NSEQ + n) * DH + d];
  const float* vb = v + bh * NSEQ * DH + d;
#pragma unroll
  for (int t = 0; t < KW; ++t) {
    const int nn = n + t - (KW / 2);
    if (nn >= 0 && nn < NSEQ) acc += cw[h * KW + t] * vb[(size_t)nn * DH];
  }
  outc[((size_t)(b * NSEQ + n)) * INNER + h * DH + d] = acc;
}

// ---------------------------------------------------------------------------
// Output projection: y = x + outc @ w_out^T + b_out
// ---------------------------------------------------------------------------
__global__ void __launch_bounds__(256, 1)
out_proj_kernel(const float* __restrict__ oc, const float* __restrict__ wout,
                const float* __restrict__ bout, const float* __restrict__ x,
                float* __restrict__ y) {
  const int wave = threadIdx.x >> 5;
  const long long tile = (long long)blockIdx.x * 8 + wave;
  const int tilesN = DIM / 64;                   // 8
  if (tile >= (long long)(ROWS / 64) * tilesN) return;
  const int ty = (int)(tile / tilesN);
  const int tx = (int)(tile % tilesN);
  ACC_INIT64;
  wave_tile_gemm64<true>(oc, INNER, wout, INNER, ty * 64, tx * 64, INNER, c);
  const int lane = threadIdx.x & 31, l15 = lane & 15, hi = lane >> 4;
  float bj[4];
#pragma unroll
  for (int j = 0; j < 4; ++j) bj[j] = bout[tx * 64 + l15 + 16 * j];
#pragma unroll
  for (int i = 0; i < 4; ++i) {
#pragma unroll
    for (int rr = 0; rr < 8; ++rr) {
      const size_t ro = (size_t)(ty * 64 + 16 * i + rr + 8 * hi) * DIM + tx * 64 + l15;
#pragma unroll
      for (int j = 0; j < 4; ++j)
        y[ro + 16 * j] = c[i][j][rr] + bj[j] + x[ro + 16 * j];
    }
  }
}

// ---------------------------------------------------------------------------
// Host orchestration
// ---------------------------------------------------------------------------
extern "C" void kernel_launch(void* const* d_in, const int* in_sizes, int n_in,
                              void* d_out, int out_size, void* d_ws, size_t ws_size,
                              hipStream_t stream) {
  const float* x     = (const float*)d_in[0];
  const float* ln_g  = (const float*)d_in[1];
  const float* ln_b  = (const float*)d_in[2];
  const float* w_qkv = (const float*)d_in[3];
  const float* w_out = (const float*)d_in[4];
  const float* b_out = (const float*)d_in[5];
  const float* cw    = (const float*)d_in[6];
  float* y = (float*)d_out;

  // Workspace layout (floats)
  float* w = (float*)d_ws;
  size_t o = 0;
  float* xn    = w + o; o += (size_t)ROWS * DIM;       // 8.39M  (reused as outc)
  float* q     = w + o; o += (size_t)BH * NSEQ * DH;   // 8.39M
  float* k     = w + o; o += (size_t)BH * NSEQ * DH;   // 8.39M  (reused as outh)
  float* v     = w + o; o += (size_t)BH * NSEQ * DH;   // 8.39M
  float* ql    = w + o; o += (size_t)BH * LM * DH;
  float* kl    = w + o; o += (size_t)BH * LM * DH;
  float* attn2 = w + o; o += (size_t)BH * LM * LM;
  float* z0    = w + o; o += (size_t)BH * LM * LM;
  float* z1    = w + o; o += (size_t)BH * LM * LM;
  float* xz    = w + o; o += (size_t)BH * LM * LM;
  float* t0    = w + o; o += (size_t)BH * LM * LM;
  float* t1    = w + o; o += (size_t)BH * LM * LM;
  float* kv    = w + o; o += (size_t)BH * LM * DH;
  float* w2    = w + o; o += (size_t)BH * LM * DH;
  float* part  = w + o; o += 64;
  float* scal  = w + o; o += 64;
  float* big   = w + o; o += (size_t)BH * NSEQ * LM;   // 33.55M (sim3 then sim1)

  const long long sQ  = (long long)NSEQ * DH;     // 262144
  const long long sL  = (long long)LM * DH;       // 16384
  const long long sS  = (long long)LM * LM;       // 65536
  const long long sB1 = (long long)NSEQ * LM;     // 1048576

  auto blocks = [](long long tiles) { return (unsigned)((tiles + 7) / 8); };

  // 1. LayerNorm
  ln_kernel<<<ROWS, 256, 0, stream>>>(x, ln_g, ln_b, xn);

  // 2. QKV projection -> q,k,v in [b,h,n,d], q pre-scaled
  qkv_gemm_kernel<<<blocks((ROWS / 64) * (3 * INNER / 64)), 256, 0, stream>>>(xn, w_qkv, q, k, v);

  // 3. Landmarks
  landmark_kernel<<<BH * LM, 64, 0, stream>>>(q, k, ql, kl);

  // 4. attn2 = softmax(ql @ kl^T)
  gemm_abt_kernel<<<dim3(blocks((LM / 64) * (LM / 64)), BH), 256, 0, stream>>>(
      ql, kl, attn2, LM, LM, DH, sL, sL, sS);
  softmax_rows_kernel<<<BH * LM, 256, 0, stream>>>(attn2, LM);

  // 5. Moore-Penrose init: z = attn2^T / (max rowsum * max colsum)
  pinv_scale_partial<<<BH, 256, 0, stream>>>(attn2, part);
  pinv_scale_final<<<1, 64, 0, stream>>>(part, scal);
  pinv_init_kernel<<<(unsigned)(BH * sS / 256), 256, 0, stream>>>(attn2, scal, z0);

  // 6. 6 Newton-Schulz iterations: z = 0.25 z (13I - xz(15I - xz(7I - xz)))
  float* zc = z0; float* zn = z1;
  const dim3 g256(blocks((LM / 64) * (LM / 64)), BH);
  for (int it = 0; it < 6; ++it) {
    gemm_nn_kernel<<<g256, 256, 0, stream>>>(attn2, zc, xz, LM, LM, LM, sS, sS, sS, 1.f, 0.f);
    axpy_diag_kernel<<<(unsigned)(BH * sS / 256), 256, 0, stream>>>(xz, t0, 7.f);
    gemm_nn_kernel<<<g256, 256, 0, stream>>>(xz, t0, t1, LM, LM, LM, sS, sS, sS, -1.f, 15.f);
    gemm_nn_kernel<<<g256, 256, 0, stream>>>(xz, t1, t0, LM, LM, LM, sS, sS, sS, -1.f, 13.f);
    gemm_nn_kernel<<<g256, 256, 0, stream>>>(zc, t0, zn, LM, LM, LM, sS, sS, sS, 0.25f, 0.f);
    float* tmp = zc; zc = zn; zn = tmp;
  }

  // 7. attn3 = softmax(ql @ k^T); kv = attn3 @ v   [bh,256,64]
  gemm_abt_kernel<<<dim3(blocks((LM / 64) * (NSEQ / 64)), BH), 256, 0, stream>>>(
      ql, k, big, LM, NSEQ, DH, sL, sQ, sB1);
  softmax_rows_kernel<<<BH * LM, 256, 0, stream>>>(big, NSEQ);
  gemm_nn_kernel<<<dim3(blocks((LM / 64) * (DH / 64)), BH), 256, 0, stream>>>(
      big, v, kv, LM, DH, NSEQ, sB1, sQ, sL, 1.f, 0.f);

  // 8. w2 = z @ kv   [bh,256,64]  (re-associated chain)
  gemm_nn_kernel<<<dim3(blocks((LM / 64) * (DH / 64)), BH), 256, 0, stream>>>(
      zc, kv, w2, LM, DH, LM, sS, sL, sL, 1.f, 0.f);

  // 9. attn1 = softmax(q @ kl^T); outh = attn1 @ w2   (outh reuses k buffer)
  gemm_abt_kernel<<<dim3(blocks((NSEQ / 64) * (LM / 64)), BH), 256, 0, stream>>>(
      q, kl, big, NSEQ, LM, DH, sQ, sL, sB1);
  softmax_rows_kernel<<<BH * NSEQ, 256, 0, stream>>>(big, LM);
  float* outh = k;  // k no longer needed
  gemm_nn_kernel<<<dim3(blocks((NSEQ / 64) * (DH / 64)), BH), 256, 0, stream>>>(
      big, w2, outh, NSEQ, DH, LM, sB1, sL, sQ, 1.f, 0.f);

  // 10. depthwise conv residual + head transpose (outc reuses xn buffer)
  float* outc = xn;
  conv_add_transpose_kernel<<<(unsigned)(BH * NSEQ / 4), 256, 0, stream>>>(outh, v, cw, outc);

  // 11. final projection + bias + residual
  out_proj_kernel<<<blocks((ROWS / 64) * (DIM / 64)), 256, 0, stream>>>(outc, w_out, b_out, x, y);
}